// CliffordSirenLayer_31817117728927
// MI455X (gfx1250) — compile-verified
//
#include <hip/hip_runtime.h>
#include <math.h>

typedef __attribute__((ext_vector_type(2))) float v2f;
typedef __attribute__((ext_vector_type(8))) float v8f;

#define B_    4
#define N_    8192
#define M_    512
#define IN_   32
#define OUT_  32
#define K_    256          // IN*8  (contraction dim, (i,c) flattened)
#define NC_   256          // OUT*8 (output dim, (o,k) flattened)
#define ROWS_ (B_ * N_)    // 32768

// ---------------------------------------------------------------------------
// Kernel 1: fold Cl(3,0) geometric-product table into weight.
// W2 logical: [kidx = i*8+c][col = o*8+k], stored pre-swizzled for WMMA B
// fragments: W2p[kb][h][col][v] with kidx = 4*kb + 2*h + v, so that a lane's
// (K, K+1) pair at a fixed column is one contiguous b64 load.
// ---------------------------------------------------------------------------
__global__ void prep_w2_kernel(const float* __restrict__ weight,
                               float* __restrict__ W2p) {
    // basis order [1,e1,e2,e3,e12,e13,e23,e123] -> masks
    const int MASK[8] = {0, 1, 2, 4, 3, 5, 6, 7};
    const int IDX[8]  = {0, 1, 2, 4, 3, 5, 6, 7};   // idx[mask] -> component

    int t = threadIdx.x;          // 1024 threads: one (o,i) pair each
    int o = t >> 5;
    int i = t & 31;

    #pragma unroll
    for (int d = 0; d < 8; ++d) {
        float w = weight[(o * IN_ + i) * 8 + d];
        if (w != w) w = 0.0f;                                   // nan_to_num
        else if (isinf(w)) w = copysignf(3.4028234663852886e38f, w);
        int bm = MASK[d];
        #pragma unroll
        for (int c = 0; c < 8; ++c) {
            int am = MASK[c];
            int k  = IDX[am ^ bm];
            int s  = 1, aa = am >> 1;
            while (aa) {
                if (__popc(aa & bm) & 1) s = -s;
                aa >>= 1;
            }
            int kidx = i * 8 + c;
            int col  = o * 8 + k;
            int kb = kidx >> 2, h = (kidx >> 1) & 1, v = kidx & 1;
            W2p[(((kb * 2 + h) * NC_) + col) * 2 + v] = (s > 0) ? w : -w;
        }
    }
}

// ---------------------------------------------------------------------------
// Kernel 2: omega[b,n] = 30*(1 + clip(MLP(qc),0,5) * exp(-min_dist)).
// Atoms for this b staged in LDS; min over squared distances (sqrt is
// monotone so min commutes), floor 1e-4 applied after the min.
// ---------------------------------------------------------------------------
__global__ void omega_kernel(const float* __restrict__ qc,
                             const float* __restrict__ ac,
                             const float* __restrict__ fw1,
                             const float* __restrict__ fb1,
                             const float* __restrict__ fw2,
                             const float* __restrict__ fb2,
                             float* __restrict__ omega) {
    __shared__ float sax[M_], say[M_], saz[M_];
    const int b = blockIdx.y;
    const int n = blockIdx.x * blockDim.x + threadIdx.x;

    for (int m = threadIdx.x; m < M_; m += blockDim.x) {
        sax[m] = ac[(b * M_ + m) * 3 + 0];
        say[m] = ac[(b * M_ + m) * 3 + 1];
        saz[m] = ac[(b * M_ + m) * 3 + 2];
    }
    __syncthreads();

    const float qx = qc[(b * N_ + n) * 3 + 0];
    const float qy = qc[(b * N_ + n) * 3 + 1];
    const float qz = qc[(b * N_ + n) * 3 + 2];

    float d2min = 3.4028234663852886e38f;
    #pragma unroll 4
    for (int m = 0; m < M_; ++m) {
        float dx = qx - sax[m], dy = qy - say[m], dz = qz - saz[m];
        float d2 = fmaf(dx, dx, fmaf(dy, dy, dz * dz));
        d2min = fminf(d2min, d2);
    }
    float mind = sqrtf(fmaxf(d2min, 1e-4f));

    float acc2 = fb2[0];
    #pragma unroll
    for (int j = 0; j < 16; ++j) {
        float z = fmaf(fw1[j * 3 + 0], qx,
                  fmaf(fw1[j * 3 + 1], qy,
                  fmaf(fw1[j * 3 + 2], qz, fb1[j])));
        float sp = fmaxf(z, 0.0f) + log1pf(__expf(-fabsf(z)));  // softplus
        acc2 = fmaf(fw2[j], sp, acc2);
    }
    float ls = fminf(fmaxf(acc2, 0.0f), 5.0f);  // clip; fmaxf(NaN,0)=0 == nan_to_num
    omega[b * N_ + n] = 30.0f * (1.0f + ls * __expf(-mind));
}

// ---------------------------------------------------------------------------
// Kernel 3: GEMM [32768 x 256] * [256 x 256] via v_wmma_f32_16x16x4_f32,
// fused bias + sin(omega * pre) epilogue.
// Block = 256 threads (8 wave32), tile = 64 rows x 256 cols.
// Each wave owns a 32-col strip across all 64 rows:
//   4 row-tiles x 2 col-tiles = 8 accumulators; per k-step only 2 global
//   b64 B-loads and 4 LDS A-loads feed 8 wmma (B reuse x4, A reuse x2).
// A fragment (32-bit 16x4 layout): lane L -> row L%16; VGPR0 holds
// K = kk + 2*(L/16), VGPR1 holds K+1  => contiguous pair per lane.
// B fragment mirrored; W2p layout makes it one contiguous b64 per lane.
// C/D layout: VGPR j -> row j + 8*(L/16), col = L%16.
// ---------------------------------------------------------------------------
__global__ void __launch_bounds__(256)
clifford_siren_main(const float* __restrict__ x,
                    const float* __restrict__ W2p,
                    const float* __restrict__ bias,
                    const float* __restrict__ omega,
                    float* __restrict__ out) {
    __shared__ float sA[64 * K_];   // 64 KB fp32 x-tile

    {   // cooperative coalesced stage of the 64x256 A tile
        const float4* src = (const float4*)(x + (size_t)blockIdx.x * 64 * K_);
        float4* dst = (float4*)sA;
        #pragma unroll
        for (int i = 0; i < 16; ++i)
            dst[threadIdx.x + i * 256] = src[threadIdx.x + i * 256];
    }
    __syncthreads();

    const int wave = threadIdx.x >> 5;   // 0..7: 32-col strip
    const int lane = threadIdx.x & 31;
    const int l15  = lane & 15;
    const int half = lane >> 4;          // 0..1

    const float* aBase = &sA[l15 * K_ + half * 2];
    const float* bBase = W2p + (half * NC_ + wave * 32 + l15) * 2;

    v8f acc[4][2] = {};

    #pragma unroll 2
    for (int kb = 0; kb < 64; ++kb) {                 // kk = 4*kb
        const float* bp = bBase + kb * (2 * NC_ * 2); // +1024 floats per kb
        v2f b0 = *(const v2f*)(bp);                   // global_load_b64 (L2 hit)
        v2f b1 = *(const v2f*)(bp + 32);              // col-tile +16
        #pragma unroll
        for (int rt = 0; rt < 4; ++rt) {
            v2f a = *(const v2f*)(aBase + rt * 16 * K_ + kb * 4); // ds b64
            acc[rt][0] = __builtin_amdgcn_wmma_f32_16x16x4_f32(
                false, a, false, b0, (short)0, acc[rt][0], false, false);
            acc[rt][1] = __builtin_amdgcn_wmma_f32_16x16x4_f32(
                false, a, false, b1, (short)0, acc[rt][1], false, false);
        }
    }

    const int col0 = wave * 32 + l15;
    const float bv0 = bias[col0];
    const float bv1 = bias[col0 + 16];

    #pragma unroll
    for (int rt = 0; rt < 4; ++rt) {
        const int row0 = blockIdx.x * 64 + rt * 16 + half * 8;
        #pragma unroll
        for (int j = 0; j < 8; ++j) {
            const float om = omega[row0 + j];
            float* orow = out + (size_t)(row0 + j) * NC_;
            orow[col0]      = __sinf(om * (acc[rt][0][j] + bv0));
            orow[col0 + 16] = __sinf(om * (acc[rt][1][j] + bv1));
        }
    }
}

// ---------------------------------------------------------------------------
extern "C" void kernel_launch(void* const* d_in, const int* in_sizes, int n_in,
                              void* d_out, int out_size, void* d_ws, size_t ws_size,
                              hipStream_t stream) {
    const float* x    = (const float*)d_in[0];
    const float* qc   = (const float*)d_in[1];
    const float* ac   = (const float*)d_in[2];
    const float* w    = (const float*)d_in[3];
    const float* bias = (const float*)d_in[4];
    const float* fw1  = (const float*)d_in[5];
    const float* fb1  = (const float*)d_in[6];
    const float* fw2  = (const float*)d_in[7];
    const float* fb2  = (const float*)d_in[8];
    float* out = (float*)d_out;

    float* ws    = (float*)d_ws;
    float* W2p   = ws;            // 256*256 floats = 256 KB
    float* omega = ws + K_ * NC_; // 32768 floats = 128 KB

    prep_w2_kernel<<<1, 1024, 0, stream>>>(w, W2p);
    omega_kernel<<<dim3(N_ / 256, B_), 256, 0, stream>>>(qc, ac, fw1, fb1,
                                                         fw2, fb2, omega);
    clifford_siren_main<<<ROWS_ / 64, 256, 0, stream>>>(x, W2p, bias, omega, out);
}